// GNNEncoder_86698209837564
// MI455X (gfx1250) — compile-verified
//
#include <hip/hip_runtime.h>
#include <hip/hip_bf16.h>

typedef __attribute__((ext_vector_type(2))) float v2f;
typedef __attribute__((ext_vector_type(8))) float v8f;

// ---------------------------------------------------------------------------
// proj16: out[row0+0..15, 0..127] = X[16 x K] @ W[K x 128] + b, no activation.
// One block = 256 threads = 8 waves; wave w owns the 16x16 tile at cols 16w.
// fp32 WMMA 16x16x4, K in {32, 64}. (~3.3 GFLOP total: negligible, keep simple)
// ---------------------------------------------------------------------------
__global__ void proj16(const float* __restrict__ X, const float* __restrict__ W,
                       const float* __restrict__ b, float* __restrict__ out,
                       int K) {
  const int lane = threadIdx.x & 31;
  const int wave = threadIdx.x >> 5;     // 0..7 -> column tile
  const int l15  = lane & 15;
  const int half = lane >> 4;            // 0 or 1 (selects K pair)
  const int row0 = blockIdx.x * 16;
  const int col  = wave * 16 + l15;      // B/D column this lane owns
  const int arow = row0 + l15;           // A row this lane loads

  v8f acc;
  const float bb = b[col];
#pragma unroll
  for (int v = 0; v < 8; ++v) acc[v] = bb;

  for (int k0 = 0; k0 < K; k0 += 4) {
    const int kb = k0 + 2 * half;
    v2f a, bv;
    a.x  = X[(size_t)arow * K + kb];
    a.y  = X[(size_t)arow * K + kb + 1];
    bv.x = W[(size_t)kb * 128 + col];
    bv.y = W[(size_t)(kb + 1) * 128 + col];
    acc = __builtin_amdgcn_wmma_f32_16x16x4_f32(false, a, false, bv,
                                                (short)0, acc, false, false);
  }

#pragma unroll
  for (int v = 0; v < 8; ++v) {
    const int r = row0 + v + 8 * half;
    out[(size_t)r * 128 + col] = acc[v];
  }
}

// ---------------------------------------------------------------------------
// scatter_mean_accum: one wave per edge; lane handles 4 features (float4).
// agg[dst] += feat[src]; cnt[dst] += 1.  (agg/cnt pre-zeroed)
// ---------------------------------------------------------------------------
__global__ void scatter_mean_accum(const float* __restrict__ feat,
                                   const int* __restrict__ src,
                                   const int* __restrict__ dst,
                                   float* __restrict__ agg,
                                   float* __restrict__ cnt, int E) {
  const int e = blockIdx.x * (blockDim.x >> 5) + (threadIdx.x >> 5);
  if (e >= E) return;
  const int lane = threadIdx.x & 31;
  const int s = src[e];
  const int d = dst[e];
  const float4 v = *(const float4*)(feat + (size_t)s * 128 + lane * 4);
  float* p = agg + (size_t)d * 128 + lane * 4;
  atomicAdd(p + 0, v.x);
  atomicAdd(p + 1, v.y);
  atomicAdd(p + 2, v.z);
  atomicAdd(p + 3, v.w);
  if (lane == 0) atomicAdd(cnt + d, 1.0f);
}

// ---------------------------------------------------------------------------
// conv32: out = relu( (Agg/max(cnt,1)) @ Wl + bl + Xin @ Wr ), H = 128.
// Block = 256 threads = 8 waves; 32 rows x 128 cols per block.
// Each wave owns one 16-wide column tile and TWO row tiles (two accumulators),
// so every B register pair loaded from Wl/Wr feeds two WMMAs (halves B
// traffic, doubles WMMA per VMEM load). A/X tiles staged in LDS with stride
// 132 (bank-conflict free for the strided A reads); staging + barrier also
// makes in-place (out == Xin) safe.
// ---------------------------------------------------------------------------
__global__ void conv32(const float* __restrict__ Agg, const float* __restrict__ cnt,
                       const float* __restrict__ Wl,  const float* __restrict__ bl,
                       const float* __restrict__ Xin, const float* __restrict__ Wr,
                       float* __restrict__ out) {
  __shared__ float As[32 * 132];
  __shared__ float Xs[32 * 132];

  const int tid  = threadIdx.x;
  const int row0 = blockIdx.x * 32;

  // Cooperative staging: mean-divide fused into the Agg load. 4096 elems/tile.
  for (int idx = tid; idx < 32 * 128; idx += 256) {
    const int r = idx >> 7;
    const int k = idx & 127;
    const float c = cnt[row0 + r];
    const float sc = 1.0f / fmaxf(c, 1.0f);
    As[r * 132 + k] = Agg[(size_t)(row0 + r) * 128 + k] * sc;
    Xs[r * 132 + k] = Xin[(size_t)(row0 + r) * 128 + k];
  }
  __syncthreads();

  const int lane = tid & 31;
  const int wave = tid >> 5;
  const int l15  = lane & 15;
  const int half = lane >> 4;
  const int col  = wave * 16 + l15;

  v8f acc0, acc1;                       // rows row0+0..15 and row0+16..31
  const float bb = bl[col];
#pragma unroll
  for (int v = 0; v < 8; ++v) { acc0[v] = bb; acc1[v] = bb; }

  // acc += (Agg/cnt) @ Wl
#pragma unroll 4
  for (int k0 = 0; k0 < 128; k0 += 4) {
    const int kb = k0 + 2 * half;
    v2f bv, a0, a1;
    bv.x = Wl[(size_t)kb * 128 + col];
    bv.y = Wl[(size_t)(kb + 1) * 128 + col];
    a0.x = As[l15 * 132 + kb];
    a0.y = As[l15 * 132 + kb + 1];
    a1.x = As[(16 + l15) * 132 + kb];
    a1.y = As[(16 + l15) * 132 + kb + 1];
    acc0 = __builtin_amdgcn_wmma_f32_16x16x4_f32(false, a0, false, bv,
                                                 (short)0, acc0, false, false);
    acc1 = __builtin_amdgcn_wmma_f32_16x16x4_f32(false, a1, false, bv,
                                                 (short)0, acc1, false, false);
  }
  // acc += Xin @ Wr
#pragma unroll 4
  for (int k0 = 0; k0 < 128; k0 += 4) {
    const int kb = k0 + 2 * half;
    v2f bv, a0, a1;
    bv.x = Wr[(size_t)kb * 128 + col];
    bv.y = Wr[(size_t)(kb + 1) * 128 + col];
    a0.x = Xs[l15 * 132 + kb];
    a0.y = Xs[l15 * 132 + kb + 1];
    a1.x = Xs[(16 + l15) * 132 + kb];
    a1.y = Xs[(16 + l15) * 132 + kb + 1];
    acc0 = __builtin_amdgcn_wmma_f32_16x16x4_f32(false, a0, false, bv,
                                                 (short)0, acc0, false, false);
    acc1 = __builtin_amdgcn_wmma_f32_16x16x4_f32(false, a1, false, bv,
                                                 (short)0, acc1, false, false);
  }

#pragma unroll
  for (int v = 0; v < 8; ++v) {
    const int r = row0 + v + 8 * half;
    out[(size_t)r * 128 + col]        = fmaxf(acc0[v], 0.0f);
    out[(size_t)(r + 16) * 128 + col] = fmaxf(acc1[v], 0.0f);
  }
}

// ---------------------------------------------------------------------------
// Host-side orchestration (graph-capture safe: only async ops on `stream`).
// Input order: x_user, x_item, user_lin_w, user_lin_b, item_lin_w, item_lin_b,
//              conv_wl, conv_bl, conv_wr, src_u2i, dst_u2i, src_i2u, dst_i2u
// Output: concat(u [NU,128], i [NI,128]) as float32.
// ---------------------------------------------------------------------------
extern "C" void kernel_launch(void* const* d_in, const int* in_sizes, int n_in,
                              void* d_out, int out_size, void* d_ws, size_t ws_size,
                              hipStream_t stream) {
  const float* x_user  = (const float*)d_in[0];
  const float* x_item  = (const float*)d_in[1];
  const float* Wu      = (const float*)d_in[2];
  const float* bu      = (const float*)d_in[3];
  const float* Wi      = (const float*)d_in[4];
  const float* bi      = (const float*)d_in[5];
  const float* conv_wl = (const float*)d_in[6];
  const float* conv_bl = (const float*)d_in[7];
  const float* conv_wr = (const float*)d_in[8];
  const int* src_u2i   = (const int*)d_in[9];
  const int* dst_u2i   = (const int*)d_in[10];
  const int* src_i2u   = (const int*)d_in[11];
  const int* dst_i2u   = (const int*)d_in[12];

  const int H  = in_sizes[3];            // 128
  const int NU = in_sizes[0] / 32;       // 200000 (divisible by 32)
  const int NI = in_sizes[1] / 64;       // 100000 (divisible by 32)
  const int E  = in_sizes[9];            // 1000000
  const int L  = in_sizes[6] / (2 * H * H);  // 2

  // Workspace layout (floats): u_buf | i_buf | agg_u | agg_i | cnt_u | cnt_i
  float* ws = (float*)d_ws;
  size_t o = 0;
  float* u_buf = ws + o; o += (size_t)NU * H;
  float* i_buf = ws + o; o += (size_t)NI * H;
  float* agg_u = ws + o; o += (size_t)NU * H;
  float* agg_i = ws + o; o += (size_t)NI * H;
  float* cnt_u = ws + o; o += (size_t)NU;
  float* cnt_i = ws + o; o += (size_t)NI;

  // Input projections.
  proj16<<<NU / 16, 256, 0, stream>>>(x_user, Wu, bu, u_buf, 32);
  proj16<<<NI / 16, 256, 0, stream>>>(x_item, Wi, bi, i_buf, 64);

  float* out_u_final = (float*)d_out;
  float* out_i_final = (float*)d_out + (size_t)NU * H;

  for (int l = 0; l < L; ++l) {
    // Zero agg_u|agg_i|cnt_u|cnt_i in one contiguous async memset.
    hipMemsetAsync(agg_u, 0,
                   ((size_t)(NU + NI) * H + (size_t)NU + (size_t)NI) * sizeof(float),
                   stream);

    // Aggregations: one wave per edge (8 edges per 256-thread block).
    scatter_mean_accum<<<(E + 7) / 8, 256, 0, stream>>>(u_buf, src_u2i, dst_u2i,
                                                        agg_i, cnt_i, E);
    scatter_mean_accum<<<(E + 7) / 8, 256, 0, stream>>>(i_buf, src_i2u, dst_i2u,
                                                        agg_u, cnt_u, E);

    float* oi = (l == L - 1) ? out_i_final : i_buf;
    float* ou = (l == L - 1) ? out_u_final : u_buf;

    // new_i = relu(mean(agg_i) @ Wl[l,0] + bl[l,0] + i @ Wr[l,0])
    conv32<<<NI / 32, 256, 0, stream>>>(
        agg_i, cnt_i,
        conv_wl + (size_t)(l * 2 + 0) * H * H, conv_bl + (size_t)(l * 2 + 0) * H,
        i_buf,  conv_wr + (size_t)(l * 2 + 0) * H * H, oi);

    // new_u = relu(mean(agg_u) @ Wl[l,1] + bl[l,1] + u @ Wr[l,1])
    conv32<<<NU / 32, 256, 0, stream>>>(
        agg_u, cnt_u,
        conv_wl + (size_t)(l * 2 + 1) * H * H, conv_bl + (size_t)(l * 2 + 1) * H,
        u_buf,  conv_wr + (size_t)(l * 2 + 1) * H * H, ou);
  }
}